// SOMLayer_9955734192298
// MI455X (gfx1250) — compile-verified
//
#include <hip/hip_runtime.h>

typedef float v2f __attribute__((ext_vector_type(2)));
typedef float v8f __attribute__((ext_vector_type(8)));

#define NODES   4096
#define FEATS   256
#define BATCH   1024
#define SIGMA_F 70.4f          // max(64,64) * 1.1
#define LR0     0.5f
#define TOT_IT  100000.0f

// Raw 32-bit LDS byte offset of a __shared__ object (addrspace(3) ptrtoint).
__device__ __forceinline__ unsigned lds_addr_of(void* p) {
    return (unsigned)(unsigned long long)(
        (__attribute__((address_space(3))) void*)p);
}

// ---------------------------------------------------------------------------
// Kernel A: w2[n] = sum_d W[n,d]^2.  One wave per node, shuffle reduce.
// ---------------------------------------------------------------------------
__global__ __launch_bounds__(256) void som_w2(const float* __restrict__ w,
                                              float* __restrict__ w2) {
    int wave = threadIdx.x >> 5, lane = threadIdx.x & 31;
    int n = blockIdx.x * 8 + wave;
    const float* row = w + n * FEATS;
    float s = 0.f;
#pragma unroll
    for (int i = 0; i < FEATS / 32; ++i) { float v = row[lane + i * 32]; s += v * v; }
#pragma unroll
    for (int off = 16; off > 0; off >>= 1) s += __shfl_xor(s, off, 32);
    if (lane == 0) w2[n] = s;
}

// ---------------------------------------------------------------------------
// Kernel B: fused  (w2[n] - 2 x.W^T)  +  per-row partial argmin.
// Block = 8 waves covering 16 batch rows x 128 nodes.  The shared 16x256 x
// tile is async-staged into LDS once per block (global_load_async_to_lds,
// ASYNCcnt) instead of each wave re-reading it; each wave then runs a
// 16x16 tile with v_wmma_f32_16x16x4_f32 over K=256 (A from LDS, B global).
// ---------------------------------------------------------------------------
__global__ __launch_bounds__(256) void som_dist_argmin(
    const float* __restrict__ x, const float* __restrict__ w,
    const float* __restrict__ w2, float* __restrict__ pval,
    int* __restrict__ pidx) {
    __shared__ float xs[16 * 260];          // 16 x 256 x-tile (+pad: 16B-aligned,
                                            //  banks spread 4*row+col)
    __shared__ float st[16 * 132];          // 16 batch rows x 128 nodes (+pad)
    const int t = threadIdx.x;
    const int wave = t >> 5, lane = t & 31;
    const int btile = blockIdx.y;           // 0..63  (16 batch rows each)
    const int ctile = blockIdx.x;           // 0..31  (128 nodes each)

    // ---- async-stage x tile: 1024 float4, 4 per thread ----
#pragma unroll
    for (int i = 0; i < 4; ++i) {
        int f   = t + i * 256;              // float4 index within tile
        int row = f >> 6;                   // 64 float4 per row
        int c4  = f & 63;
        unsigned la = lds_addr_of(&xs[row * 260 + c4 * 4]);
        unsigned long long ga =
            (unsigned long long)(x + (size_t)(btile * 16 + row) * FEATS + c4 * 4);
        asm volatile("global_load_async_to_lds_b128 %0, %1, off"
                     :: "v"(la), "v"(ga) : "memory");
    }
    asm volatile("s_wait_asynccnt 0" ::: "memory");
    __syncthreads();

    const int n0 = ctile * 128 + wave * 16;
    const int mrow = lane & 15;
    const int ksub = (lane >> 4) * 2;       // K split across lane halves
    const float* wrow = w + (size_t)(n0 + mrow) * FEATS;
    const float* arow = &xs[mrow * 260];

    v8f acc = {};
#pragma unroll 4
    for (int k = 0; k < FEATS; k += 4) {
        v2f a = *(const v2f*)(arow + k + ksub);   // A[m,k] = x[b, k]  (LDS)
        v2f b = *(const v2f*)(wrow + k + ksub);   // B[k,n] = W[n, k]  (global)
        acc = __builtin_amdgcn_wmma_f32_16x16x4_f32(
            false, a, false, b, (short)0, acc, false, false);
    }

    const float w2v = w2[n0 + mrow];
    const int hi = (lane >> 4) * 8;
    const int col = wave * 16 + mrow;
#pragma unroll
    for (int r = 0; r < 8; ++r)
        st[(r + hi) * 132 + col] = w2v - 2.0f * acc[r];
    __syncthreads();

    // argmin over this block's 128 node columns: 16 threads per batch row
    const int row = t >> 4, j = t & 15;
    float bv = 3.4e38f; int bc = 0;
#pragma unroll
    for (int i = 0; i < 8; ++i) {
        int c = j + i * 16;
        float v = st[row * 132 + c];
        if (v < bv) { bv = v; bc = c; }
    }
#pragma unroll
    for (int off = 8; off > 0; off >>= 1) {
        float ov = __shfl_xor(bv, off, 16);
        int   oc = __shfl_xor(bc, off, 16);
        if (ov < bv || (ov == bv && oc < bc)) { bv = ov; bc = oc; }
    }
    if (j == 0) {
        int b = btile * 16 + row;
        pval[b * 32 + ctile] = bv;
        pidx[b * 32 + ctile] = ctile * 128 + bc;
    }
}

// ---------------------------------------------------------------------------
// Kernel C: reduce 32 partials per batch row -> bmu index.
// ---------------------------------------------------------------------------
__global__ __launch_bounds__(256) void som_argmin_final(
    const float* __restrict__ pval, const int* __restrict__ pidx,
    int* __restrict__ bmu, float* __restrict__ out_idx) {
    int t = threadIdx.x;
    int b = blockIdx.x * 8 + (t >> 5);
    int lane = t & 31;
    float v = pval[b * 32 + lane];
    int   i = pidx[b * 32 + lane];
#pragma unroll
    for (int off = 16; off > 0; off >>= 1) {
        float ov = __shfl_xor(v, off, 32);
        int   oi = __shfl_xor(i, off, 32);
        if (ov < v || (ov == v && oi < i)) { v = ov; i = oi; }
    }
    if (lane == 0) { bmu[b] = i; out_idx[b] = (float)i; }
}

// ---------------------------------------------------------------------------
// Kernel D: update GEMM  U = L^T x  with L generated on the fly in LDS.
// One block (4 waves) per 16-node tile; each wave owns 64 feature columns
// (4 accumulators).  ds_add_f32 atomics accumulate S[n] = sum_b L[b,n].
// Next batch chunk of x is prefetched (global_prefetch_b8) while computing.
// Epilogue: out = W + (U - S*W)/B.
// ---------------------------------------------------------------------------
__global__ __launch_bounds__(128) void som_update(
    const float* __restrict__ x, const float* __restrict__ w,
    const int* __restrict__ bmu, const int* __restrict__ itp,
    float* __restrict__ out) {
    __shared__ float lr_chunk[16 * 66];     // 16 nodes x 64 batch (+pad)
    __shared__ float s_sum[16];
    const int t = threadIdx.x;
    const int wave = t >> 5, lane = t & 31;
    const int nt = blockIdx.x;              // node tile, 0..255

    const float itf    = (float)(*itp);
    const float decay  = __expf(-itf * (SIGMA_F / TOT_IT));
    const float radius = SIGMA_F * decay + 1e-6f;
    const float lr = LR0 * decay;
    const float r2 = radius * radius;
    const float inv2r2 = 0.5f / r2;

    if (t < 16) s_sum[t] = 0.f;

    const int mrow = lane & 15;
    const int ksub = (lane >> 4) * 2;
    const int fb = wave * 64 + mrow;        // feature base for this wave/lane
    v8f acc0 = {}, acc1 = {}, acc2 = {}, acc3 = {};

    for (int cc = 0; cc < BATCH / 64; ++cc) {
        __syncthreads();                    // previous chunk fully consumed
        // ---- prefetch next chunk's x block (64 rows x 1KB = 512 lines) ----
        if (cc + 1 < BATCH / 64) {
            const float* nx = x + (size_t)((cc + 1) * 64) * FEATS;
#pragma unroll
            for (int i = 0; i < 4; ++i)
                __builtin_prefetch(nx + (size_t)(t * 4 + i) * 32, 0, 3);
        }
        // ---- generate lr_op tile: 16 nodes x 64 batch, 8 values/thread ----
#pragma unroll
        for (int i = 0; i < 8; ++i) {
            int v    = t + i * 128;
            int node = v >> 6;
            int bb   = v & 63;
            int bidx = bmu[cc * 64 + bb];
            int g    = nt * 16 + node;
            float di = (float)((g >> 6) - (bidx >> 6));
            float dj = (float)((g & 63) - (bidx & 63));
            float d2 = di * di + dj * dj;
            float val = (d2 <= r2) ? lr * __expf(-d2 * inv2r2) : 0.0f;
            lr_chunk[node * 66 + bb] = val;
            atomicAdd(&s_sum[node], val);   // ds_add_f32
        }
        __syncthreads();
        // ---- WMMA: A = L^T (from LDS), B = x chunk (global, L2-hot) ----
        const float* xbase = x + (size_t)(cc * 64 + ksub) * FEATS;
#pragma unroll 4
        for (int kk = 0; kk < 64; kk += 4) {
            v2f a = *(const v2f*)(&lr_chunk[mrow * 66 + kk + ksub]);
            const float* xp = xbase + (size_t)kk * FEATS;
            v2f b0 = { xp[fb],      xp[FEATS + fb]      };
            v2f b1 = { xp[fb + 16], xp[FEATS + fb + 16] };
            v2f b2 = { xp[fb + 32], xp[FEATS + fb + 32] };
            v2f b3 = { xp[fb + 48], xp[FEATS + fb + 48] };
            acc0 = __builtin_amdgcn_wmma_f32_16x16x4_f32(false, a, false, b0, (short)0, acc0, false, false);
            acc1 = __builtin_amdgcn_wmma_f32_16x16x4_f32(false, a, false, b1, (short)0, acc1, false, false);
            acc2 = __builtin_amdgcn_wmma_f32_16x16x4_f32(false, a, false, b2, (short)0, acc2, false, false);
            acc3 = __builtin_amdgcn_wmma_f32_16x16x4_f32(false, a, false, b3, (short)0, acc3, false, false);
        }
    }
    __syncthreads();                        // s_sum complete

    const int hi = (lane >> 4) * 8;
    const float invB = 1.0f / (float)BATCH;
#pragma unroll
    for (int r = 0; r < 8; ++r) {
        int g = nt * 16 + r + hi;
        float ss = s_sum[r + hi];
        const float* wr = w + (size_t)g * FEATS;
        float* orow = out + (size_t)g * FEATS;
        int c = wave * 64 + mrow;
        float w0 = wr[c];       orow[c]      = w0  + (acc0[r] - ss * w0)  * invB;
        float w1 = wr[c + 16];  orow[c + 16] = w1  + (acc1[r] - ss * w1)  * invB;
        float w2_ = wr[c + 32]; orow[c + 32] = w2_ + (acc2[r] - ss * w2_) * invB;
        float w3 = wr[c + 48];  orow[c + 48] = w3  + (acc3[r] - ss * w3)  * invB;
    }
}

// ---------------------------------------------------------------------------
extern "C" void kernel_launch(void* const* d_in, const int* in_sizes, int n_in,
                              void* d_out, int out_size, void* d_ws, size_t ws_size,
                              hipStream_t stream) {
    const float* x  = (const float*)d_in[0];   // (1024, 256)
    const float* w  = (const float*)d_in[1];   // (4096, 256)
    const int*  itp = (const int*)d_in[2];     // scalar iteration
    float* out = (float*)d_out;                // [new_map (4096*256) | bmu (1024)]

    float* ws   = (float*)d_ws;
    float* w2   = ws;                          // 4096
    float* pval = ws + 4096;                   // 1024*32
    int*   pidx = (int*)(ws + 4096 + 32768);   // 1024*32
    int*   bmu  = (int*)(ws + 4096 + 65536);   // 1024

    som_w2<<<NODES / 8, 256, 0, stream>>>(w, w2);
    som_dist_argmin<<<dim3(32, 64), 256, 0, stream>>>(x, w, w2, pval, pidx);
    som_argmin_final<<<BATCH / 8, 256, 0, stream>>>(pval, pidx, bmu,
                                                    out + (size_t)NODES * FEATS);
    som_update<<<NODES / 16, 128, 0, stream>>>(x, w, bmu, itp, out);
}